// CTRGC_10075993276448
// MI455X (gfx1250) — compile-verified
//
#include <hip/hip_runtime.h>
#include <hip/hip_bf16.h>
#include <math.h>
#include <stdint.h>

// Shapes
#define NN 256
#define VV 25
#define TT 128
#define CIN 64
#define COUT 64
#define REL 9
#define H4 18
#define H5 6

typedef __attribute__((ext_vector_type(2))) float v2f;
typedef __attribute__((ext_vector_type(8))) float v8f;
typedef __attribute__((ext_vector_type(4))) unsigned int v4u;
typedef __attribute__((ext_vector_type(4))) int v4i;
typedef __attribute__((ext_vector_type(8))) int v8i;

__device__ __forceinline__ float silu(float x) {
    return x / (1.f + __expf(-x));
}

// -------------------------------------------------------------------------
// TDM: 1-row tile of `ndw` DWORDs global->LDS.  pad=1 inserts 1 DWORD of LDS
// padding after every 64 DWORDs (pad_interval code 5, pad_amount code 0),
// i.e. hardware-produced stride-65 rows for bank-conflict-free columns.
// -------------------------------------------------------------------------
__device__ __forceinline__ void tdm_load_1d(uint32_t lds_off, const void* gaddr,
                                            uint32_t ndw, int pad)
{
    uint64_t ga = (uint64_t)(uintptr_t)gaddr;
    v4u g0;
    g0.x = 1u;                                           // count=1 (valid), user mode
    g0.y = lds_off;                                      // lds_addr (bytes)
    g0.z = (uint32_t)ga;                                 // global_addr[31:0]
    g0.w = (uint32_t)((ga >> 32) & 0x1FFFFFFu) | (2u << 30);  // addr[56:32] | type=2

    uint32_t w0 = (2u << 16);                            // data_size = 4B
    if (pad) w0 |= (1u << 20) | (5u << 22);              // pad_enable, 64 DW interval, 1 DW pad
    v8i g1;
    g1[0] = (int)w0;
    g1[1] = (int)((ndw & 0xFFFFu) << 16);                // tensor_dim0 lo16 (atomic addr = 0)
    g1[2] = (int)(((ndw >> 16) & 0xFFFFu) | (1u << 16)); // tensor_dim0 hi16 | tensor_dim1 = 1
    g1[3] = (int)(ndw << 16);                            // tile_dim0 = ndw (fits 16b)
    g1[4] = 1;                                           // tile_dim1 = 1, tile_dim2 = 0
    g1[5] = (int)ndw;                                    // tensor_dim0_stride lo32
    g1[6] = 0;
    g1[7] = 0;

    v4i gz = { 0, 0, 0, 0 };
#if __clang_major__ >= 23
    v8i gz8 = { 0, 0, 0, 0, 0, 0, 0, 0 };
    __builtin_amdgcn_tensor_load_to_lds(g0, g1, gz, gz, gz8, 0);
#else
    __builtin_amdgcn_tensor_load_to_lds(g0, g1, gz, gz, 0);
#endif
}

// -------------------------------------------------------------------------
// Kernel 1: per (n,v) tile. TDM stages the 128x64 x-tile into stride-65 LDS;
// computes xm = mean_T(x) and x3 = x@W3 + b3 via V_WMMA_F32_16X16X4_F32.
// x3 stored as (N, 9, V, T).
// -------------------------------------------------------------------------
__global__ __launch_bounds__(256) void proj_kernel(
    const float* __restrict__ x, const float* __restrict__ W3,
    const float* __restrict__ b3, float* __restrict__ xm,
    float* __restrict__ x3)
{
    __shared__ float xs[TT * 65];    // 128x64 tile, stride-65 (TDM pad-on-load)
    __shared__ float wpad[CIN * 16]; // W3 padded 9->16 cols
    __shared__ float bpad[16];

    const int tid  = threadIdx.x;
    const int wave = tid >> 5, lane = tid & 31;
    const int nv   = blockIdx.x;          // n*25 + v
    const int n    = nv / VV, v = nv % VV;
    const float* gx = x + (size_t)nv * (TT * CIN);

    // Each wave DMAs its 16-row (1024 DWORD) slice; LDS slice starts at
    // wave*16*65*4 bytes; padding keeps rows at stride 65 floats.
    tdm_load_1d((uint32_t)(uintptr_t)xs + (uint32_t)wave * (16 * 65 * 4),
                gx + wave * (16 * CIN), 16 * CIN, /*pad=*/1);

    // overlap: small weight loads through the normal path
    for (int i = tid; i < CIN * 16; i += 256) {
        int r = i >> 4, c = i & 15;
        wpad[i] = (c < REL) ? W3[r * REL + c] : 0.f;
    }
    if (tid < 16) bpad[tid] = (tid < REL) ? b3[tid] : 0.f;

    __builtin_amdgcn_s_wait_tensorcnt(0);
    __syncthreads();

    // temporal mean -> xm (N,V,64)
    if (tid < CIN) {
        float s = 0.f;
        for (int t = 0; t < TT; ++t) s += xs[t * 65 + tid];
        xm[(size_t)nv * CIN + tid] = s * (1.f / 128.f);
    }

    // WMMA: each wave handles one 16-row M-tile; K=64 in 16 steps of 4
    const int half = lane >> 4, col = lane & 15;
    const int t0 = wave * 16;
    const int arow = t0 + col;           // A-matrix row for this lane

    const float bb = bpad[col];
    v8f acc = { bb, bb, bb, bb, bb, bb, bb, bb };

    for (int k0 = 0; k0 < CIN; k0 += 4) {
        int k = k0 + 2 * half;
        v2f a, b;
        a.x = xs[arow * 65 + k];
        a.y = xs[arow * 65 + k + 1];
        b.x = wpad[k * 16 + col];
        b.y = wpad[(k + 1) * 16 + col];
        acc = __builtin_amdgcn_wmma_f32_16x16x4_f32(
            false, a, false, b, (short)0, acc, false, false);
    }

    if (col < REL) {
        size_t base = (((size_t)n * REL + col) * VV + v) * TT;
        #pragma unroll
        for (int j = 0; j < 8; ++j) {
            int t = t0 + j + 8 * half;
            x3[base + t] = acc[j];
        }
    }
}

// -------------------------------------------------------------------------
// Kernel 2: per n. m1/m2 = xm@W{1,2}+b; pairwise rel-MLP (SiLU); +A.
// h stored as (N, 9, 25, 25).
// -------------------------------------------------------------------------
__global__ __launch_bounds__(256) void rel_kernel(
    const float* __restrict__ xm, const float* __restrict__ A,
    const float* __restrict__ W1, const float* __restrict__ b1,
    const float* __restrict__ W2, const float* __restrict__ b2,
    const float* __restrict__ W4a, const float* __restrict__ b4a,
    const float* __restrict__ W4b, const float* __restrict__ b4b,
    float* __restrict__ hws)
{
    __shared__ float xmL[VV * CIN];
    __shared__ float m1L[VV * REL], m2L[VV * REL];
    __shared__ float W1L[CIN * REL], W2L[CIN * REL];
    __shared__ float W4aL[REL * H4], W4bL[H4 * REL];
    __shared__ float b4aL[H4], b4bL[REL], b1L[REL], b2L[REL];
    __shared__ float AL[VV * VV];

    const int tid = threadIdx.x, n = blockIdx.x;

    for (int i = tid; i < VV * CIN; i += 256) xmL[i] = xm[(size_t)n * VV * CIN + i];
    for (int i = tid; i < CIN * REL; i += 256) { W1L[i] = W1[i]; W2L[i] = W2[i]; }
    for (int i = tid; i < REL * H4; i += 256) { W4aL[i] = W4a[i]; W4bL[i] = W4b[i]; }
    for (int i = tid; i < VV * VV; i += 256) AL[i] = A[i];
    if (tid < H4) b4aL[tid] = b4a[tid];
    if (tid < REL) { b4bL[tid] = b4b[tid]; b1L[tid] = b1[tid]; b2L[tid] = b2[tid]; }
    __syncthreads();

    for (int i2 = tid; i2 < 2 * VV * REL; i2 += 256) {
        int which = i2 / (VV * REL), i = i2 % (VV * REL);
        int r = i / REL, c = i % REL;
        const float* W = which ? W2L : W1L;
        float s = which ? b2L[c] : b1L[c];
        for (int k = 0; k < CIN; ++k) s += xmL[r * CIN + k] * W[k * REL + c];
        (which ? m2L : m1L)[i] = s;
    }
    __syncthreads();

    for (int p = tid; p < VV * VV; p += 256) {
        int u = p / VV, vv = p % VV;
        float rel[REL], hh[H4];
        #pragma unroll
        for (int c = 0; c < REL; ++c) rel[c] = m1L[u * REL + c] - m2L[vv * REL + c];
        #pragma unroll
        for (int j = 0; j < H4; ++j) {
            float s = b4aL[j];
            #pragma unroll
            for (int c = 0; c < REL; ++c) s += rel[c] * W4aL[c * H4 + j];
            hh[j] = silu(s);
        }
        float av = AL[p];
        #pragma unroll
        for (int c = 0; c < REL; ++c) {
            float s = b4bL[c];
            #pragma unroll
            for (int j = 0; j < H4; ++j) s += hh[j] * W4bL[j * REL + c];
            hws[(((size_t)n * REL + c) * VV + u) * VV + vv] = silu(s) + av;
        }
    }
}

// -------------------------------------------------------------------------
// Kernel 3: per n. TDM stages the full x3[n] slab (115 KB) into LDS;
// agg[u,t,c] = sum_v h[c,u,v]*x3[c,v,t]; y6 = silu(agg@W5a+b);
// out = silu(y6@W5b+b) via WMMA f32 16x16x4 (K padded 6->8, 4 col tiles).
// -------------------------------------------------------------------------
__global__ __launch_bounds__(256) void agg_out_kernel(
    const float* __restrict__ x3ws, const float* __restrict__ hws,
    const float* __restrict__ W5a, const float* __restrict__ b5a,
    const float* __restrict__ W5b, const float* __restrict__ b5b,
    float* __restrict__ out)
{
    __shared__ float x3L[REL * VV * TT];   // 28800 f (115.2 KB)
    __shared__ float hL[REL * VV * VV];    // 5625 f
    __shared__ float W5aL[REL * H5];
    __shared__ float b5aL[H5];
    __shared__ float W5bL[8 * COUT];       // rows padded 6->8
    __shared__ float b5bL[COUT];
    __shared__ float ypad[8][32][8];       // per-wave A staging (K padded)

    const int tid = threadIdx.x, n = blockIdx.x;
    const int wave = tid >> 5, lane = tid & 31;

    // Each wave DMAs a contiguous 3600-DWORD slice of the x3[n] slab.
    const float* gx3 = x3ws + (size_t)n * (REL * VV * TT);
    tdm_load_1d((uint32_t)(uintptr_t)x3L + (uint32_t)wave * (3600 * 4),
                gx3 + wave * 3600, 3600, /*pad=*/0);

    // overlap: small tables through the normal path
    for (int i = tid; i < REL * VV * VV; i += 256) hL[i] = hws[(size_t)n * REL * VV * VV + i];
    if (tid < REL * H5) W5aL[tid] = W5a[tid];
    if (tid < H5) b5aL[tid] = b5a[tid];
    for (int i = tid; i < 8 * COUT; i += 256) {
        int r = i >> 6, c = i & 63;
        W5bL[i] = (r < H5) ? W5b[r * COUT + c] : 0.f;
    }
    if (tid < COUT) b5bL[tid] = b5b[tid];

    __builtin_amdgcn_s_wait_tensorcnt(0);
    __syncthreads();

    const int half = lane >> 4, col = lane & 15;

    // 100 chunks of 32 positions; chunk ch -> u = ch/4, t in [(ch%4)*32, +32)
    for (int ch = wave; ch < 100; ch += 8) {
        const int u = ch >> 2, t0 = (ch & 3) * 32;
        const int t = t0 + lane;

        float agg[REL];
        #pragma unroll
        for (int c = 0; c < REL; ++c) {
            float s = 0.f;
            const float* hp = &hL[(c * VV + u) * VV];    // broadcast across lanes
            const float* xp = &x3L[c * VV * TT + t];     // conflict-free
            #pragma unroll
            for (int v = 0; v < VV; ++v) s += hp[v] * xp[v * TT];
            agg[c] = s;
        }
        #pragma unroll
        for (int j = 0; j < H5; ++j) {
            float s = b5aL[j];
            #pragma unroll
            for (int c = 0; c < REL; ++c) s += agg[c] * W5aL[c * H5 + j];
            ypad[wave][lane][j] = silu(s);
        }
        ypad[wave][lane][6] = 0.f;
        ypad[wave][lane][7] = 0.f;
        asm volatile("s_wait_dscnt 0x0" ::: "memory");   // intra-wave LDS RAW

        const size_t obase = (((size_t)n * VV + u) * TT) * (size_t)COUT;
        #pragma unroll
        for (int r = 0; r < 2; ++r) {
            const int arow = r * 16 + col;
            #pragma unroll
            for (int ct = 0; ct < 4; ++ct) {
                const int ocol = ct * 16 + col;
                const float bb = b5bL[ocol];
                v8f acc = { bb, bb, bb, bb, bb, bb, bb, bb };
                #pragma unroll
                for (int s4 = 0; s4 < 8; s4 += 4) {
                    int k = s4 + 2 * half;
                    v2f a, b;
                    a.x = ypad[wave][arow][k];
                    a.y = ypad[wave][arow][k + 1];
                    b.x = W5bL[k * COUT + ocol];
                    b.y = W5bL[(k + 1) * COUT + ocol];
                    acc = __builtin_amdgcn_wmma_f32_16x16x4_f32(
                        false, a, false, b, (short)0, acc, false, false);
                }
                #pragma unroll
                for (int j = 0; j < 8; ++j) {
                    int trow = t0 + r * 16 + j + 8 * half;
                    out[obase + (size_t)trow * COUT + ocol] = silu(acc[j]);
                }
            }
        }
    }
}

// -------------------------------------------------------------------------
extern "C" void kernel_launch(void* const* d_in, const int* in_sizes, int n_in,
                              void* d_out, int out_size, void* d_ws, size_t ws_size,
                              hipStream_t stream) {
    const float* x   = (const float*)d_in[0];
    const float* A   = (const float*)d_in[1];
    const float* W1  = (const float*)d_in[2];
    const float* b1  = (const float*)d_in[3];
    const float* W2  = (const float*)d_in[4];
    const float* b2  = (const float*)d_in[5];
    const float* W3  = (const float*)d_in[6];
    const float* b3  = (const float*)d_in[7];
    const float* W4a = (const float*)d_in[8];
    const float* b4a = (const float*)d_in[9];
    const float* W4b = (const float*)d_in[10];
    const float* b4b = (const float*)d_in[11];
    const float* W5a = (const float*)d_in[12];
    const float* b5a = (const float*)d_in[13];
    const float* W5b = (const float*)d_in[14];
    const float* b5b = (const float*)d_in[15];
    float* out = (float*)d_out;

    float* ws  = (float*)d_ws;
    float* xm  = ws;                                   // 256*25*64   = 409600
    float* x3  = ws + (size_t)NN * VV * CIN;           // 256*9*25*128= 7372800
    float* hws = x3 + (size_t)NN * REL * VV * TT;      // 256*9*25*25 = 1440000

    proj_kernel<<<dim3(NN * VV), dim3(256), 0, stream>>>(x, W3, b3, xm, x3);
    rel_kernel<<<dim3(NN), dim3(256), 0, stream>>>(xm, A, W1, b1, W2, b2,
                                                   W4a, b4a, W4b, b4b, hws);
    agg_out_kernel<<<dim3(NN), dim3(256), 0, stream>>>(x3, hws, W5a, b5a,
                                                       W5b, b5b, out);
}